// MatrixExp_14465449853582
// MI455X (gfx1250) — compile-verified
//
#include <hip/hip_runtime.h>

// ---------------------------------------------------------------------------
// out = X @ exp(S),  exp(S) = 9-term Taylor of the antisymmetric unfold of
// S_flat.  Q = exp(S) built via 8 square f32 WMMA GEMMs (Q-update fused in
// the epilogue), then one 8192x2048x2048 GEMM for X@Q.
// Matrix math: V_WMMA_F32_16X16X4_F32 (full f32).
// Tile staging: CDNA5 async global->LDS DMA (GLOBAL_LOAD_ASYNC_TO_LDS_B128),
// double-buffered, one barrier per K-panel.  Wave tile 64x64 (16 acc tiles)
// for high FLOP-per-LDS-byte intensity.
// ---------------------------------------------------------------------------

typedef __attribute__((ext_vector_type(2))) float v2f;
typedef __attribute__((ext_vector_type(8))) float v8f;
typedef __attribute__((address_space(3))) void lds_void;

#define NDIM 2048
#define BM 128
#define BN 256
#define BK 32
#define LDA_S (BK + 4)   // 36 floats/row = 144 B: 16B-aligned, conflict-free
#define LDB_S (BN + 4)   // 260 floats/row = 1040 B: 16B-aligned

// 32-bit LDS byte offset (as(3) ptrtoint) for async-DMA destination VGPR.
__device__ __forceinline__ unsigned lds_addr_u32(void* p) {
  return (unsigned)(unsigned long long)(lds_void*)p;
}

__device__ __forceinline__ void async_load_b128(void* lds_dst, const void* gsrc) {
  asm volatile("global_load_async_to_lds_b128 %0, %1, off"
               :
               : "v"(lds_addr_u32(lds_dst)), "v"((unsigned long long)gsrc)
               : "memory");
}

__device__ __forceinline__ void wait_async0() {
  asm volatile("s_wait_asynccnt 0x0" ::: "memory");
}

// ---------------------------------------------------------------------------
// Scatter strict upper triangle of S_flat, antisymmetrize:
//   S[i][j] = S_flat[tri(i,j)] (j>i) ; -S_flat[tri(j,i)] (j<i) ; 0 (i==j)
// Also initializes T = S and Q = I + S (first two Taylor terms).
// tri(r,c) = r*N - r*(r+1)/2 + (c - r - 1)   (row-major triu_indices order)
// ---------------------------------------------------------------------------
__global__ __launch_bounds__(256)
void init_S_kernel(const float* __restrict__ s_flat,
                   float* __restrict__ S,
                   float* __restrict__ T,
                   float* __restrict__ Q) {
  const int idx = blockIdx.x * 256 + threadIdx.x;
  const int i = idx / NDIM;
  const int j = idx - i * NDIM;
  float v = 0.0f;
  if (j > i) {
    v = s_flat[i * NDIM - ((i * (i + 1)) >> 1) + (j - i - 1)];
  } else if (j < i) {
    v = -s_flat[j * NDIM - ((j * (j + 1)) >> 1) + (i - j - 1)];
  }
  S[idx] = v;
  T[idx] = v;
  Q[idx] = (i == j) ? (v + 1.0f) : v;
}

// ---------------------------------------------------------------------------
// D = alpha * (A @ B), optionally Qacc += D.
// A: MxK row-major, B: KxN row-major, K == N == NDIM.
// Block tile 128x256, K-step 32, 8 waves: 2(M) x 4(N), wave tile 64x64.
// Async-DMA double-buffered LDS pipeline, one barrier per K-panel.
// ---------------------------------------------------------------------------
__global__ __launch_bounds__(256)
void gemm_f32_wmma(const float* __restrict__ A,
                   const float* __restrict__ B,
                   float* __restrict__ D,
                   float* __restrict__ Qacc,
                   float alpha) {
  __shared__ float lds_a[2][BM * LDA_S];
  __shared__ float lds_b[2][BK * LDB_S];

  const int t     = threadIdx.x;
  const int lane  = t & 31;
  const int wave  = t >> 5;
  const int wm0   = (wave & 1) * 64;   // wave M offset inside block tile
  const int wn0   = (wave >> 1) * 64;  // wave N offset inside block tile
  const int bm    = blockIdx.y * BM;
  const int bn    = blockIdx.x * BN;
  const int l15   = lane & 15;
  const int khalf = (lane >> 4) * 2;   // lane half selects K pair {0,1}/{2,3}

  // per-thread staging coordinates (16 B per lane per instruction)
  const int ar = t >> 3;          // A rows ar + 32*i, col ac
  const int ac = (t & 7) * 4;
  const int br = t >> 6;          // B rows br + 4*i, col bc
  const int bc = (t & 63) * 4;

  v8f acc[4][4] = {};             // 16 x 16x16 f32 accumulator tiles

  // Issue async DMA for the K-panel starting at k0 into buffer `buf`.
  auto issue_tile = [&](int k0, int buf) {
#pragma unroll
    for (int i = 0; i < 4; ++i) {
      const int r = ar + i * 32;
      async_load_b128(&lds_a[buf][r * LDA_S + ac],
                      A + (size_t)(bm + r) * NDIM + (k0 + ac));
    }
#pragma unroll
    for (int i = 0; i < 8; ++i) {
      const int r = br + i * 4;
      async_load_b128(&lds_b[buf][r * LDB_S + bc],
                      B + (size_t)(k0 + r) * NDIM + (bn + bc));
    }
  };

  // 8 K-steps of 4, 16 WMMAs each, from lds[cur].
  auto compute_panel = [&](int cur) {
#pragma unroll
    for (int kk = 0; kk < BK; kk += 4) {
      v2f a[4], b[4];
#pragma unroll
      for (int mi = 0; mi < 4; ++mi) {
        const int row = wm0 + mi * 16 + l15;      // A frag: M=lane&15, K=2g+v
        a[mi].x = lds_a[cur][row * LDA_S + kk + khalf];
        a[mi].y = lds_a[cur][row * LDA_S + kk + khalf + 1];
      }
#pragma unroll
      for (int ni = 0; ni < 4; ++ni) {
        const int col = wn0 + ni * 16 + l15;      // B frag: N=lane&15, K=2g+v
        b[ni].x = lds_b[cur][(kk + khalf) * LDB_S + col];
        b[ni].y = lds_b[cur][(kk + khalf + 1) * LDB_S + col];
      }
#pragma unroll
      for (int mi = 0; mi < 4; ++mi)
#pragma unroll
        for (int ni = 0; ni < 4; ++ni)
          acc[mi][ni] = __builtin_amdgcn_wmma_f32_16x16x4_f32(
              false, a[mi], false, b[ni], (short)0, acc[mi][ni], false, false);
    }
  };

  const int KT = NDIM / BK;       // 64 K-panels

  issue_tile(0, 0);
  wait_async0();
  __syncthreads();

  // steady state: branch-free DMA issue for kt+1 under compute of kt
  for (int kt = 0; kt < KT - 1; ++kt) {
    const int cur = kt & 1;
    issue_tile((kt + 1) * BK, cur ^ 1);
    compute_panel(cur);
    wait_async0();     // this wave's DMA into the other buffer has landed
    __syncthreads();   // everyone done reading cur / writing nxt
  }
  compute_panel((KT - 1) & 1);   // peeled last panel

  // ---- epilogue: C/D layout M = 8*(lane>>4)+v, N = lane&15 ----
#pragma unroll
  for (int mi = 0; mi < 4; ++mi) {
#pragma unroll
    for (int ni = 0; ni < 4; ++ni) {
      const int n = bn + wn0 + ni * 16 + l15;
#pragma unroll
      for (int v = 0; v < 8; ++v) {
        const int m = bm + wm0 + mi * 16 + (lane >> 4) * 8 + v;
        const float val = acc[mi][ni][v] * alpha;
        const size_t off = (size_t)m * NDIM + n;
        if (Qacc) Qacc[off] += val;   // uniform branch, fused Taylor update
        D[off] = val;
      }
    }
  }
}

// ---------------------------------------------------------------------------
extern "C" void kernel_launch(void* const* d_in, const int* in_sizes, int n_in,
                              void* d_out, int out_size, void* d_ws,
                              size_t ws_size, hipStream_t stream) {
  const float* X      = (const float*)d_in[0];
  const float* S_flat = (const float*)d_in[1];
  float* out = (float*)d_out;

  const size_t mat = (size_t)NDIM * NDIM;   // 4M floats = 16 MB
  float* S  = (float*)d_ws;                 // [ 0, 16) MB
  float* T  = S + mat;                      // [16, 32) MB
  float* Tn = T + mat;                      // [32, 48) MB
  float* Q  = Tn + mat;                     // [48, 64) MB

  // Q = I + S, T = S
  init_S_kernel<<<(unsigned)(mat / 256), 256, 0, stream>>>(S_flat, S, T, Q);

  // Taylor terms 2..9:  Tn = (T @ S) / i ;  Q += Tn
  dim3 blk(256);
  dim3 grid_sq(NDIM / BN, NDIM / BM);
  for (int i = 2; i <= 9; ++i) {
    gemm_f32_wmma<<<grid_sq, blk, 0, stream>>>(T, S, Tn, Q, 1.0f / (float)i);
    float* tmp = T; T = Tn; Tn = tmp;
  }

  // out = X @ Q
  dim3 grid_x(NDIM / BN, 8192 / BM);
  gemm_f32_wmma<<<grid_x, blk, 0, stream>>>(X, Q, out, nullptr, 1.0f);
}